// SoftPolygonPyTorch_15573551415898
// MI455X (gfx1250) — compile-verified
//
#include <hip/hip_runtime.h>
#include <hip/hip_bf16.h>

typedef __attribute__((ext_vector_type(2))) float v2f;
typedef __attribute__((ext_vector_type(8))) float v8f;

#define NVERT 32          // vertices (== edges) per polygon
#define IMG_H 128
#define IMG_W 128
#define ROWS_PER_BLOCK 8  // one image row per wave, 8 waves / block (256 thr)

// D = A(16x4) * B(4x16) + 0, f32 WMMA. K rows of B: [gx, gy, 1, gx^2+gy^2].
// Five affine fields per 16-edge x 16-pixel tile:
//   dot   : (dx/sq, dy/sq, -(x1*dx+y1*dy)/sq, 0)
//   dotF  : dot coefficients scaled by F = sq+1e-5  (so segmid = d1 - dot*dotF)
//   d1    : (-2x1, -2y1, x1^2+y1^2, 1)   == |p-v1|^2
//   d2    : (-2x2, -2y2, x2^2+y2^2, 1)   == |p-v2|^2
//   thr-gx: (-1, slope, x1-slope*y1, 0)  crossing test: cond1 && (thr-gx > 0)

__global__ __launch_bounds__(256)
void soft_polygon_wmma(const float* __restrict__ verts,
                       float* __restrict__ out)
{
    __shared__ float s_aD[NVERT], s_bD[NVERT], s_cD[NVERT];
    __shared__ float s_aF[NVERT], s_bF[NVERT], s_cF[NVERT];
    __shared__ float s_a1[NVERT], s_b1[NVERT], s_c1[NVERT];
    __shared__ float s_a2[NVERT], s_b2[NVERT], s_c2[NVERT];
    __shared__ float s_sl[NVERT], s_tc[NVERT];
    __shared__ float s_fy[NVERT], s_ty[NVERT];

    const int tid = threadIdx.x;
    const int b   = blockIdx.x / (IMG_H / ROWS_PER_BLOCK);
    const int y0  = (blockIdx.x % (IMG_H / ROWS_PER_BLOCK)) * ROWS_PER_BLOCK;

    // ---- per-edge parameter setup (one thread per edge) ----
    if (tid < NVERT) {
        const float* vb = verts + b * NVERT * 2;
        const int e  = tid;
        const int en = (e + 1) & (NVERT - 1);
        const int ep = (e + NVERT - 1) & (NVERT - 1);
        const float x1 = vb[2*e],  y1 = vb[2*e+1];
        const float x2 = vb[2*en], y2 = vb[2*en+1];
        const float px = vb[2*ep], py = vb[2*ep+1];   // prev vertex (tx,ty)
        const float dx = x2 - x1, dy = y2 - y1;
        const float sq  = dx*dx + dy*dy + 1e-5f;
        const float inv = 1.0f / sq;
        const float F   = sq + 1e-5f;                 // segmid = d1 - dot^2*F
        const float cDv = -(x1*dx + y1*dy) * inv;
        s_aD[e] = dx*inv;      s_bD[e] = dy*inv;      s_cD[e] = cDv;
        s_aF[e] = dx*inv*F;    s_bF[e] = dy*inv*F;    s_cF[e] = cDv*F;
        s_a1[e] = -2.0f*x1;    s_b1[e] = -2.0f*y1;    s_c1[e] = x1*x1 + y1*y1;
        s_a2[e] = -2.0f*x2;    s_b2[e] = -2.0f*y2;    s_c2[e] = x2*x2 + y2*y2;
        const float sl = (px - x1) / (py - y1);       // inf/nan -> cross false, matches ref
        s_sl[e] = sl;          s_tc[e] = x1 - sl*y1;
        s_fy[e] = y1;          s_ty[e] = py;
    }
    __syncthreads();

    const int  lane = tid & 31;
    const bool lo   = lane < 16;
    const int  wv   = tid >> 5;
    const int  y    = y0 + wv;
    const float gyf = (float)y;

    // ---- build A matrices (2 edge tiles x 5 fields), batch-constant ----
    // A layout (16x4, wave32): lanes 0-15 -> M=lane,    v.x=K0, v.y=K1
    //                          lanes16-31 -> M=lane-16, v.x=K2, v.y=K3
    v2f A_dot[2], A_dF[2], A_d1[2], A_d2[2], A_th[2];
#pragma unroll
    for (int T = 0; T < 2; ++T) {
        const int e = T*16 + (lane & 15);
        A_dot[T] = lo ? (v2f){s_aD[e], s_bD[e]} : (v2f){s_cD[e], 0.0f};
        A_dF[T]  = lo ? (v2f){s_aF[e], s_bF[e]} : (v2f){s_cF[e], 0.0f};
        A_d1[T]  = lo ? (v2f){s_a1[e], s_b1[e]} : (v2f){s_c1[e], 1.0f};
        A_d2[T]  = lo ? (v2f){s_a2[e], s_b2[e]} : (v2f){s_c2[e], 1.0f};
        A_th[T]  = lo ? (v2f){-1.0f,   s_sl[e]} : (v2f){s_tc[e], 0.0f};
    }

    // cond1[e] = (fy[e] > gy) != (ty[e] > gy): gy is wave-uniform -> one ballot-xor
    const float fyv = s_fy[lane], tyv = s_ty[lane];
    const unsigned cond =
        (unsigned)(__ballot(fyv > gyf) ^ __ballot(tyv > gyf));

    float* orow = out + ((size_t)b * IMG_H + y) * IMG_W;
    const v8f zero = {};

    for (int xg = 0; xg < IMG_W / 16; ++xg) {
        const float gxf = (float)(xg*16 + (lane & 15));
        const float r2  = gxf*gxf + gyf*gyf;
        // B layout (4x16): lanes 0-15 -> col=lane, v.x=K0(gx), v.y=K1(gy)
        //                  lanes16-31 -> col=lane-16, v.x=K2(1), v.y=K3(r2)
        const v2f Bv = lo ? (v2f){gxf, gyf} : (v2f){1.0f, r2};

        float minv = 3.4e38f;
        int   cnt  = 0;
#pragma unroll
        for (int T = 0; T < 2; ++T) {
            v8f dotm = __builtin_amdgcn_wmma_f32_16x16x4_f32(false, A_dot[T], false, Bv, (short)0, zero, false, false);
            v8f dFm  = __builtin_amdgcn_wmma_f32_16x16x4_f32(false, A_dF[T],  false, Bv, (short)0, zero, false, false);
            v8f d1m  = __builtin_amdgcn_wmma_f32_16x16x4_f32(false, A_d1[T],  false, Bv, (short)0, zero, false, false);
            v8f d2m  = __builtin_amdgcn_wmma_f32_16x16x4_f32(false, A_d2[T],  false, Bv, (short)0, zero, false, false);
            v8f thm  = __builtin_amdgcn_wmma_f32_16x16x4_f32(false, A_th[T],  false, Bv, (short)0, zero, false, false);
#pragma unroll
            for (int k = 0; k < 8; ++k) {
                // C/D layout: VGPR k = edge (T*16+k) in lanes 0-15, edge (T*16+k+8) in lanes 16-31
                const float dv  = dotm[k];
                float seg = __builtin_fmaf(-dv, dFm[k], d1m[k]); // d1 - dot^2*(sq+1e-5)
                seg = (dv < 0.0f) ? d1m[k] : seg;
                seg = (dv > 1.0f) ? d2m[k] : seg;
                minv = fminf(minv, seg);
                const bool bk  = ((cond >> (T*16 + k))     & 1u) != 0u; // uniform
                const bool bk8 = ((cond >> (T*16 + k + 8)) & 1u) != 0u; // uniform
                const float cv = (lo ? bk : bk8) ? thm[k] : -1.0f;      // NaN-safe
                cnt += (cv > 0.0f) ? 1 : 0;
            }
        }
        // combine the two lane-halves (edges k vs k+8): SWAPX16
        minv = fminf(minv, __shfl_xor(minv, 16));
        cnt += __shfl_xor(cnt, 16);

        const float io  = (cnt & 1) ? 1.0f : -1.0f;
        const float arg = io * minv;
        const float res = 1.0f / (1.0f + __expf(-arg));
        if (lo) orow[xg*16 + lane] = res;
    }
}

extern "C" void kernel_launch(void* const* d_in, const int* in_sizes, int n_in,
                              void* d_out, int out_size, void* d_ws, size_t ws_size,
                              hipStream_t stream) {
    const float* verts = (const float*)d_in[0];   // (B, 32, 2) f32
    float* out = (float*)d_out;                   // (B, 128, 128) f32
    const int B = in_sizes[0] / (NVERT * 2);      // 32 for the reference setup
    dim3 grid(B * (IMG_H / ROWS_PER_BLOCK));      // 512 blocks
    dim3 block(32 * ROWS_PER_BLOCK);              // 8 waves/block
    soft_polygon_wmma<<<grid, block, 0, stream>>>(verts, out);
}